// LinearMulti_6124623364236
// MI455X (gfx1250) — compile-verified
//
#include <hip/hip_runtime.h>

#define NMODELS   128
#define SZ        256          // SZ_IN == SZ_OUT
#define B_TOTAL   4096
#define MTILE     16
#define MAX_MTILES (B_TOTAL / MTILE)   // worst case: all rows in one model
#define LDS_STRIDE (SZ + 4)            // pad to break stride-256 bank conflicts

typedef __attribute__((ext_vector_type(2))) float v2f;
typedef __attribute__((ext_vector_type(8))) float v8f;

// ---------------- counting-sort phase ----------------

__global__ void lm_zero_kernel(int* counts) {
    if (threadIdx.x < NMODELS) counts[threadIdx.x] = 0;
}

__global__ void lm_count_kernel(const int* __restrict__ ids, int* counts) {
    int i = blockIdx.x * blockDim.x + threadIdx.x;
    if (i < B_TOTAL) atomicAdd(&counts[ids[i]], 1);
}

__global__ void lm_scan_kernel(const int* __restrict__ counts,
                               int* offsets, int* cursor) {
    if (threadIdx.x == 0) {
        int s = 0;
        for (int m = 0; m < NMODELS; ++m) {
            offsets[m] = s;
            cursor[m]  = s;
            s += counts[m];
        }
    }
}

__global__ void lm_scatter_kernel(const int* __restrict__ ids,
                                  int* cursor, int* __restrict__ sorted_idx) {
    int i = blockIdx.x * blockDim.x + threadIdx.x;
    if (i < B_TOTAL) {
        int pos = atomicAdd(&cursor[ids[i]], 1);
        sorted_idx[pos] = i;
    }
}

// ---------------- grouped GEMM phase ----------------
// grid = (NMODELS, MAX_MTILES); block = 512 threads = 16 waves.
// Block (m, t): rows sorted_idx[off[m] + 16t .. +16t+15] of model m.
// Wave w owns output columns [16w, 16w+16) during compute, then stores row w.

__global__ __launch_bounds__(512)
void lm_gemm_kernel(const float* __restrict__ inp,        // [B, 256]
                    const float* __restrict__ weight_lut, // [128, 256*256] (k-major)
                    const float* __restrict__ bias_lut,   // [128, 256]
                    const int*   __restrict__ offsets,    // [128]
                    const int*   __restrict__ counts,     // [128]
                    const int*   __restrict__ sorted_idx, // [B]
                    float* __restrict__ out)              // [B, 256]
{
    const int model = blockIdx.x;
    const int mtile = blockIdx.y;
    const int count = counts[model];
    const int mbase = mtile * MTILE;
    if (mbase >= count) return;                 // uniform: whole block exits
    const int off  = offsets[model];
    const int rows = min(MTILE, count - mbase);

    __shared__ __align__(16) float ldsA[MTILE][LDS_STRIDE];
    __shared__ int ldsRow[MTILE];

    const int tid  = threadIdx.x;
    const int wave = tid >> 5;
    const int lane = tid & 31;

    // ---- stage gathered A tile (16 rows x 256) into LDS ----
    {
        const int r = wave;                     // 16 waves <-> 16 rows
        const int k = lane << 3;                // 8 floats per lane
        int idx = (r < rows) ? sorted_idx[off + mbase + r] : -1;
        if (lane == 0) ldsRow[r] = idx;
        float4 v0 = make_float4(0.f, 0.f, 0.f, 0.f);
        float4 v1 = v0;
        if (idx >= 0) {                          // wave-uniform branch
            const float* src = inp + (size_t)idx * SZ + k;
            v0 = *(const float4*)(src);
            v1 = *(const float4*)(src + 4);
        }
        *(float4*)&ldsA[r][k]     = v0;
        *(float4*)&ldsA[r][k + 4] = v1;
    }
    __syncthreads();

    // ---- per-wave 16x16 output tile via V_WMMA_F32_16X16X4_F32 ----
    const int n     = (wave << 4) + (lane & 15); // output column
    const int m     = lane & 15;                 // A row for this lane
    const int khalf = (lane >> 4) << 1;          // K sub-offset: 0 or 2
    const int mhi   = (lane >> 4) << 3;          // C/D row base: 0 or 8

    const float* Wcol = weight_lut + (size_t)model * (SZ * SZ) + n; // column n
    const float  bias = bias_lut[model * SZ + n];

    v8f c;
#pragma unroll
    for (int i = 0; i < 8; ++i) c[i] = bias;

#pragma unroll 8
    for (int k = 0; k < SZ; k += 4) {
        const int k0 = k + khalf;
        v2f a;
        a.x = ldsA[m][k0];
        a.y = ldsA[m][k0 + 1];
        v2f b;
        b.x = Wcol[(size_t)k0 * SZ];             // W[k0][n]
        b.y = Wcol[(size_t)(k0 + 1) * SZ];       // W[k0+1][n]
        // 8 args: (neg_a, A, neg_b, B, c_mod, C, reuse_a, reuse_b)
        c = __builtin_amdgcn_wmma_f32_16x16x4_f32(
                false, a, false, b, (short)0, c, false, false);
    }

    // ---- epilogue: transpose D tile through LDS, then coalesced row stores --
    __syncthreads();                 // all waves finished reading ldsA

#pragma unroll
    for (int r = 0; r < 8; ++r) {
        ldsA[r + mhi][n] = c[r];     // D reg r -> tile row (r + mhi), col n
    }
    __syncthreads();

    {
        const int r  = wave;         // wave w stores tile row w
        const int gr = ldsRow[r];
        if (gr >= 0) {               // wave-uniform branch
            const int k = lane << 3;
            float4 v0 = *(const float4*)&ldsA[r][k];
            float4 v1 = *(const float4*)&ldsA[r][k + 4];
            float* dst = out + (size_t)gr * SZ + k;
            *(float4*)(dst)     = v0;
            *(float4*)(dst + 4) = v1;
        }
    }
}

// ---------------- launch ----------------

extern "C" void kernel_launch(void* const* d_in, const int* in_sizes, int n_in,
                              void* d_out, int out_size, void* d_ws, size_t ws_size,
                              hipStream_t stream) {
    const float* inp        = (const float*)d_in[0];   // [4096, 256]
    const int*   agent_ids  = (const int*)  d_in[1];   // [4096]
    const float* weight_lut = (const float*)d_in[2];   // [128, 65536]
    const float* bias_lut   = (const float*)d_in[3];   // [128, 256]
    float* out = (float*)d_out;

    int* ws      = (int*)d_ws;
    int* counts  = ws;                 // 128
    int* offsets = ws + NMODELS;       // 128
    int* cursor  = ws + 2 * NMODELS;   // 128
    int* sorted  = ws + 3 * NMODELS;   // 4096

    lm_zero_kernel   <<<1, NMODELS, 0, stream>>>(counts);
    lm_count_kernel  <<<B_TOTAL / 256, 256, 0, stream>>>(agent_ids, counts);
    lm_scan_kernel   <<<1, 32, 0, stream>>>(counts, offsets, cursor);
    lm_scatter_kernel<<<B_TOTAL / 256, 256, 0, stream>>>(agent_ids, cursor, sorted);

    dim3 grid(NMODELS, MAX_MTILES);
    lm_gemm_kernel<<<grid, 512, 0, stream>>>(inp, weight_lut, bias_lut,
                                             offsets, counts, sorted, out);
}